// SparseGAM_67903432949823
// MI455X (gfx1250) — compile-verified
//
#include <hip/hip_runtime.h>
#include <hip/hip_bf16.h>

#define B_    32
#define T_    992
#define TOUT_ 32
#define F_    256
#define K_    16384
#define NTOT_ 1024   // T_ + TOUT_

typedef __attribute__((ext_vector_type(2))) float v2f;
typedef __attribute__((ext_vector_type(8))) float v8f;

// ---------------------------------------------------------------------------
// Kernel 1: nodes_cat = concat(nodes, x)  (written into d_out region)
// One float4 per thread; exact grid, no bounds check needed.
// ---------------------------------------------------------------------------
__global__ void concat_kernel(const float4* __restrict__ nodes,
                              const float4* __restrict__ x,
                              float4* __restrict__ out) {
    int idx = blockIdx.x * blockDim.x + threadIdx.x;   // over B*1024*64 float4
    int f4  = idx & 63;          // 256 floats = 64 float4 per row
    int row = idx >> 6;
    int b   = row >> 10;         // /1024
    int r   = row & 1023;
    float4 v;
    if (r < T_) v = nodes[((size_t)b * T_ + r) * 64 + f4];
    else        v = x[((size_t)b * TOUT_ + (r - T_)) * 64 + f4];
    out[idx] = v;
}

// ---------------------------------------------------------------------------
// Kernel 2: sequential-scan sparse message pass.
// One wave32 per (b, t). Exact cumsum rank via ballot + prefix popcount.
// ---------------------------------------------------------------------------
__global__ void msg_kernel(const float* __restrict__ nodes_cat,
                           const long long* __restrict__ edges,
                           const float* __restrict__ weights,
                           float* __restrict__ msg) {
    int b    = blockIdx.y;
    int wave = threadIdx.x >> 5;
    int lane = threadIdx.x & 31;
    int t    = blockIdx.x * 8 + wave;          // 8 waves/block, 4 blocks in x
    long long n = (long long)(T_ + t);

    const long long* srcp = edges + (size_t)b * 2 * K_;
    const long long* dstp = srcp + K_;

    float acc[8] = {0.f, 0.f, 0.f, 0.f, 0.f, 0.f, 0.f, 0.f};
    int c = 0;                                  // running cumsum(le)

    for (int k0 = 0; k0 < K_; k0 += 32) {
        long long s = srcp[k0 + lane];
        long long d = dstp[k0 + lane];
        bool le = (s <= n) && (d <= n);
        unsigned lemask = __builtin_amdgcn_ballot_w32(le);
        unsigned hmask  = __builtin_amdgcn_ballot_w32(le && (d == n));
        int si = (int)s;                        // src < 1024 fits in 32 bits
        while (hmask) {
            int hl = __ffs(hmask) - 1;          // uniform across wave
            hmask &= hmask - 1;
            // cumsum at hit = c + popc(le lanes <= hl); rank = cumsum - 1
            int rank = c + __popc(lemask & ((1u << hl) - 1u));
            int srow = __builtin_amdgcn_readlane(si, hl);
            float w  = weights[(size_t)b * K_ + rank];
            const float* xv = nodes_cat + ((size_t)b * NTOT_ + srow) * F_;
            #pragma unroll
            for (int j = 0; j < 8; ++j)
                acc[j] += w * xv[lane + 32 * j];
        }
        c += __popc(lemask);
    }

    float* mp = msg + ((size_t)b * TOUT_ + t) * F_;
    #pragma unroll
    for (int j = 0; j < 8; ++j)
        mp[lane + 32 * j] = acc[j];
}

// ---------------------------------------------------------------------------
// Kernel 3: mx = x @ W1 + msg @ W2 + bias using V_WMMA_F32_16X16X4_F32.
// M = B*TOUT = 1024, N = F = 256. One wave per 16x16 tile of C.
// A layout (16x4 f32): lane m = lane&15; VGPR0/1 hold K = kb, kb+1 with
// kb = (lane>=16 ? 2 : 0). B (4x16) mirrored; C/D row-striped (8 VGPRs).
// ---------------------------------------------------------------------------
__global__ void gemm_kernel(const float* __restrict__ x,
                            const float* __restrict__ msg,
                            const float* __restrict__ W1,
                            const float* __restrict__ W2,
                            const float* __restrict__ bias,
                            float* __restrict__ mx) {
    int wave = threadIdx.x >> 5;
    int lane = threadIdx.x & 31;
    int tile = blockIdx.x * 8 + wave;        // 64 x 16 = 1024 tiles
    int m0 = (tile >> 4) << 4;               // row tile
    int n0 = (tile & 15) << 4;               // col tile
    int ml = m0 + (lane & 15);               // A row for this lane
    int nl = n0 + (lane & 15);               // B col for this lane
    int kb = (lane >> 4) << 1;               // 0 or 2

    v8f c = {};

    const float* ap = x + (size_t)ml * F_;
    #pragma unroll 4
    for (int k = 0; k < F_; k += 4) {
        v2f a, bv;
        a.x  = ap[k + kb];
        a.y  = ap[k + kb + 1];
        bv.x = W1[(size_t)(k + kb) * F_ + nl];
        bv.y = W1[(size_t)(k + kb + 1) * F_ + nl];
        c = __builtin_amdgcn_wmma_f32_16x16x4_f32(false, a, false, bv,
                                                  (short)0, c, false, false);
    }
    const float* mp = msg + (size_t)ml * F_;
    #pragma unroll 4
    for (int k = 0; k < F_; k += 4) {
        v2f a, bv;
        a.x  = mp[k + kb];
        a.y  = mp[k + kb + 1];
        bv.x = W2[(size_t)(k + kb) * F_ + nl];
        bv.y = W2[(size_t)(k + kb + 1) * F_ + nl];
        c = __builtin_amdgcn_wmma_f32_16x16x4_f32(false, a, false, bv,
                                                  (short)0, c, false, false);
    }

    float bn = bias[nl];
    int rbase = m0 + ((lane >> 4) << 3);     // lane<16 -> M=r, else M=r+8
    int col   = n0 + (lane & 15);
    #pragma unroll
    for (int r = 0; r < 8; ++r)
        mx[(size_t)(rbase + r) * F_ + col] = c[r] + bn;
}

// ---------------------------------------------------------------------------
// Host launcher
// ---------------------------------------------------------------------------
extern "C" void kernel_launch(void* const* d_in, const int* in_sizes, int n_in,
                              void* d_out, int out_size, void* d_ws, size_t ws_size,
                              hipStream_t stream) {
    const float*     x       = (const float*)d_in[0];       // [B,TOUT,F]
    const float*     nodes   = (const float*)d_in[1];       // [B,T,F]
    const long long* edges   = (const long long*)d_in[2];   // [B,2,K] int64
    const float*     weights = (const float*)d_in[3];       // [B,K,1]
    const float*     W1      = (const float*)d_in[4];       // [F,F]
    const float*     W2      = (const float*)d_in[5];       // [F,F]
    const float*     bias    = (const float*)d_in[6];       // [F]

    // Output tuple layout (flat, in return order):
    //   mx [B*TOUT*F] f32 | nodes_cat [B*NTOT*F] f32 | edge_list bytes | weights f32
    float* mx        = (float*)d_out;
    float* nodes_cat = mx + (size_t)B_ * TOUT_ * F_;
    char*  edge_out  = (char*)(nodes_cat + (size_t)B_ * NTOT_ * F_);
    size_t edge_bytes = (size_t)B_ * 2 * K_ * sizeof(long long);
    float* w_out     = (float*)(edge_out + edge_bytes);

    float* msg = (float*)d_ws;   // [B,TOUT,F] f32 = 1 MB scratch

    // 1) nodes_cat (also the source table for the message pass)
    int n4 = (B_ * NTOT_ * F_) / 4;              // 2,097,152 float4
    concat_kernel<<<n4 / 256, 256, 0, stream>>>(
        (const float4*)nodes, (const float4*)x, (float4*)nodes_cat);

    // 2) pass-through tuple outputs
    hipMemcpyAsync(edge_out, edges, edge_bytes, hipMemcpyDeviceToDevice, stream);
    hipMemcpyAsync(w_out, weights, (size_t)B_ * K_ * sizeof(float),
                   hipMemcpyDeviceToDevice, stream);

    // 3) message pass: grid (TOUT/8, B), 8 wave32 per block
    msg_kernel<<<dim3(TOUT_ / 8, B_), 256, 0, stream>>>(
        nodes_cat, edges, weights, msg);

    // 4) WMMA GEMM: 1024 tiles / 8 waves per block = 128 blocks
    gemm_kernel<<<128, 256, 0, stream>>>(x, msg, W1, W2, bias, mx);
}